// DiffPoolLayer_65283502899366
// MI455X (gfx1250) — compile-verified
//
#include <hip/hip_runtime.h>
#include <hip/hip_bf16.h>

typedef float v2f __attribute__((ext_vector_type(2)));
typedef float v8f __attribute__((ext_vector_type(8)));

#define IN_F 64
#define OUT_F 64

// ---------------------------------------------------------------- utilities
__global__ void k_fill_zero(float* __restrict__ p, int n) {
    int i = blockIdx.x * blockDim.x + threadIdx.x;
    if (i < n) p[i] = 0.0f;
}

// deg[col[e]] += ew[e]
__global__ void k_deg(const int* __restrict__ col, const float* __restrict__ ew,
                      float* __restrict__ deg, int E) {
    int e = blockIdx.x * blockDim.x + threadIdx.x;
    if (e < E) atomicAdd(deg + col[e], ew[e]);
}

// dinv = rsqrt(deg + 1)  (self-loop weight 1; always > 0)
__global__ void k_dinv(float* __restrict__ deg, int N) {
    int i = blockIdx.x * blockDim.x + threadIdx.x;
    if (i < N) deg[i] = rsqrtf(deg[i] + 1.0f);
}

// ------------------------------------------------- h = x @ W  (WMMA fp32)
// One wave computes one 16x16 D tile. Block = 128 threads = 4 waves covering
// the 4 column tiles of OUT_F=64; grid.x = row tiles of 16.
__global__ void k_gemm_wmma(const float* __restrict__ x, const float* __restrict__ W,
                            float* __restrict__ h, int N) {
    const int lane = threadIdx.x & 31;
    const int wave = threadIdx.x >> 5;          // 0..3 -> column tile
    const int m0   = blockIdx.x * 16;
    const int n0   = wave * 16;
    const int mrow = lane & 15;                 // M (for A) / N (for B,D)
    const int khalf = (lane >> 4) * 2;          // K sub-offset per half-wave

    v8f c = {0.f,0.f,0.f,0.f,0.f,0.f,0.f,0.f};

    #pragma unroll
    for (int k = 0; k < IN_F; k += 4) {
        // A tile 16x4: lane holds x[m0+mrow, k+khalf .. k+khalf+1]
        int m = m0 + mrow; if (m >= N) m = N - 1;     // clamp (rows not stored)
        v2f a = *(const v2f*)(x + (size_t)m * IN_F + k + khalf);
        // B tile 4x16: lane holds W[k+khalf, n0+mrow], W[k+khalf+1, n0+mrow]
        v2f b;
        b.x = W[(size_t)(k + khalf)     * OUT_F + n0 + mrow];
        b.y = W[(size_t)(k + khalf + 1) * OUT_F + n0 + mrow];
        c = __builtin_amdgcn_wmma_f32_16x16x4_f32(
                /*neg_a=*/false, a, /*neg_b=*/false, b,
                /*c_mod=*/(short)0, c, /*reuse_a=*/false, /*reuse_b=*/false);
    }

    // D layout: VGPR r -> row m0 + r + 8*(lane>=16), col n0 + (lane&15)
    const int rbase = m0 + ((lane >> 4) << 3);
    const int colD  = n0 + (lane & 15);
    #pragma unroll
    for (int r = 0; r < 8; ++r) {
        int m = rbase + r;
        if (m < N) h[(size_t)m * OUT_F + colD] = c[r];
    }
}

// ------------------------------------------------- hp = x @ W_pool  [N]
// One wave per node: coalesced float2 loads + shuffle reduction (wave32).
__global__ void k_gemv_pool(const float* __restrict__ x, const float* __restrict__ Wp,
                            float* __restrict__ hp, int N) {
    int w    = (blockIdx.x * blockDim.x + threadIdx.x) >> 5;
    int lane = threadIdx.x & 31;
    if (w >= N) return;
    float2 a  = ((const float2*)(x + (size_t)w * IN_F))[lane];
    float2 wp = ((const float2*)Wp)[lane];
    float  s  = a.x * wp.x + a.y * wp.y;
    #pragma unroll
    for (int off = 16; off; off >>= 1) s += __shfl_xor(s, off, 32);
    if (lane == 0) hp[w] = s;
}

// ------------------------------------- self-loop init (non-atomic full write)
// out[i,:] = dinv[i]^2 * h[i,:] ; score[j] = dinv[j>>6]^2 * hp[j] for j < N
__global__ void k_selfloop(const float* __restrict__ dinv, const float* __restrict__ h,
                           const float* __restrict__ hp, float* __restrict__ out,
                           float* __restrict__ score, int N) {
    int idx = blockIdx.x * blockDim.x + threadIdx.x;      // over N*64
    int total = N * OUT_F;
    if (idx < total) {
        int i = idx >> 6;
        float d2 = dinv[i] * dinv[i];
        out[idx] = d2 * h[idx];
        if (idx < N) score[idx] = d2 * hp[idx];
    }
}

// ------------------------------------------------- edge scatter (atomics)
// One wave per edge: norm = dinv[r]*w*dinv[c]; 64-wide fp32 atomic adds.
__global__ void k_edge_scatter(const int* __restrict__ row, const int* __restrict__ col,
                               const float* __restrict__ ew, const float* __restrict__ dinv,
                               const float* __restrict__ h, const float* __restrict__ hp,
                               float* __restrict__ out, float* __restrict__ score, int E) {
    int e    = (blockIdx.x * blockDim.x + threadIdx.x) >> 5;
    int lane = threadIdx.x & 31;
    if (e >= E) return;
    int r = row[e], c = col[e];
    float norm = dinv[r] * ew[e] * dinv[c];
    float2 hv = ((const float2*)(h + (size_t)r * OUT_F))[lane];
    float* op = out + (size_t)c * OUT_F + lane * 2;
    atomicAdd(op,     norm * hv.x);
    atomicAdd(op + 1, norm * hv.y);
    if (lane == 0) atomicAdd(score + c, norm * hp[r]);
}

// ---------------------------------------- segment softmax (ordered-uint trick)
__device__ __forceinline__ unsigned enc_f(float f) {
    unsigned u = __float_as_uint(f);
    return (u & 0x80000000u) ? ~u : (u | 0x80000000u);
}
__device__ __forceinline__ float dec_f(unsigned u) {
    return __uint_as_float((u & 0x80000000u) ? (u & 0x7FFFFFFFu) : ~u);
}

__global__ void k_segmax(const float* __restrict__ score, const float* __restrict__ bp,
                         const int* __restrict__ seg, unsigned* __restrict__ smax, int N) {
    int i = blockIdx.x * blockDim.x + threadIdx.x;
    if (i < N) atomicMax(smax + seg[i], enc_f(score[i] + bp[0]));
}

__global__ void k_expsum(const float* __restrict__ score, const float* __restrict__ bp,
                         const int* __restrict__ seg, const unsigned* __restrict__ smax,
                         float* __restrict__ expv, float* __restrict__ z, int N) {
    int i = blockIdx.x * blockDim.x + threadIdx.x;
    if (i < N) {
        float e = expf(score[i] + bp[0] - dec_f(smax[seg[i]]));
        expv[i] = e;
        atomicAdd(z + seg[i], e);
    }
}

// ------------------------- res[g,:] += s_i * relu(out[i,:] + b_embed)
__global__ void k_pool(const float* __restrict__ out, const float* __restrict__ be,
                       const float* __restrict__ expv, const float* __restrict__ z,
                       const int* __restrict__ seg, float* __restrict__ res, int N) {
    int i    = (blockIdx.x * blockDim.x + threadIdx.x) >> 5;
    int lane = threadIdx.x & 31;
    if (i >= N) return;
    int g = seg[i];
    float s = expv[i] / z[g];
    float2 v = ((const float2*)(out + (size_t)i * OUT_F))[lane];
    float2 b = ((const float2*)be)[lane];
    float e0 = fmaxf(v.x + b.x, 0.0f);
    float e1 = fmaxf(v.y + b.y, 0.0f);
    float* rp = res + (size_t)g * OUT_F + lane * 2;
    atomicAdd(rp,     s * e0);
    atomicAdd(rp + 1, s * e1);
}

// ---------------------------------------------------------------- launcher
extern "C" void kernel_launch(void* const* d_in, const int* in_sizes, int n_in,
                              void* d_out, int out_size, void* d_ws, size_t ws_size,
                              hipStream_t stream) {
    const float* x   = (const float*)d_in[0];
    const float* We  = (const float*)d_in[1];
    const float* be  = (const float*)d_in[2];
    const float* Wp  = (const float*)d_in[3];
    const float* bp  = (const float*)d_in[4];
    const float* ew  = (const float*)d_in[5];
    const int*   ei  = (const int*)d_in[6];
    const int*   seg = (const int*)d_in[7];
    float* res = (float*)d_out;

    const int N = in_sizes[0] / IN_F;
    const int E = in_sizes[5];
    const int G = out_size / OUT_F;
    const int* row = ei;          // edge_index[0]
    const int* col = ei + E;      // edge_index[1]

    // workspace layout (floats)
    float* h     = (float*)d_ws;                 // N*64
    float* out   = h + (size_t)N * OUT_F;        // N*64
    float* deg   = out + (size_t)N * OUT_F;      // N  (becomes dinv)
    float* hp    = deg + N;                      // N
    float* score = hp + N;                       // N
    float* expv  = score + N;                    // N
    unsigned* smax = (unsigned*)(expv + N);      // G
    float* z     = (float*)(smax + G);           // G

    const int T = 256;
    // 1) zero accumulators
    k_fill_zero<<<(N + T - 1) / T, T, 0, stream>>>(deg, N);
    k_fill_zero<<<(2 * G + T - 1) / T, T, 0, stream>>>((float*)smax, 2 * G);
    k_fill_zero<<<(out_size + T - 1) / T, T, 0, stream>>>(res, out_size);
    // 2) degree + inverse-sqrt norm
    k_deg<<<(E + T - 1) / T, T, 0, stream>>>(col, ew, deg, E);
    k_dinv<<<(N + T - 1) / T, T, 0, stream>>>(deg, N);
    // 3) dense projections
    k_gemm_wmma<<<(N + 15) / 16, 128, 0, stream>>>(x, We, h, N);
    k_gemv_pool<<<(N + 7) / 8, T, 0, stream>>>(x, Wp, hp, N);
    // 4) self-loop init, then edge scatter
    k_selfloop<<<((size_t)N * OUT_F + T - 1) / T, T, 0, stream>>>(deg, h, hp, out, score, N);
    k_edge_scatter<<<(E + 7) / 8, T, 0, stream>>>(row, col, ew, deg, h, hp, out, score, E);
    // 5) per-graph softmax + weighted pooling
    k_segmax<<<(N + T - 1) / T, T, 0, stream>>>(score, bp, seg, smax, N);
    k_expsum<<<(N + T - 1) / T, T, 0, stream>>>(score, bp, seg, smax, expv, z, N);
    k_pool<<<(N + 7) / 8, T, 0, stream>>>(out, be, expv, z, seg, res, N);
}